// SPD_GCN_73882027426331
// MI455X (gfx1250) — compile-verified
//
#include <hip/hip_runtime.h>

// ---------------------------------------------------------------------------
// SPD-GCN for MI455X (gfx1250), fp32, wave32.
// Memory-bound (~400 MB traffic vs ~3.5 GFLOP @ 23.3 TB/s => ~17us floor):
// keep fp32, two passes over x (second pass mostly L2-resident: 134 MB of x
// < 192 MB global L2). Matrix work via V_WMMA_F32_16X16X4_F32; x tile staged
// into LDS with GLOBAL_LOAD_ASYNC_TO_LDS_B128 (ASYNCcnt-tracked async path).
// ---------------------------------------------------------------------------

typedef __attribute__((ext_vector_type(2))) float v2f;
typedef __attribute__((ext_vector_type(8))) float v8f;

#define BATCH   32
#define CCH     256
#define LPIX    4096
#define NNODE   16
#define MIDD    4
#define NCHUNK  16      // pass-1 pixel chunks per batch
#define CHUNKPX 256
#define TILEPX  64      // pass-3 pixels per block
#define XPAD    72      // x-tile row pitch: 288B rows -> 16B aligned for B128
                        // async LDS writes; row-pair bank shift of 8 keeps the
                        // two half-wave B-fragment reads on disjoint banks
#define TPAD    273     // theta1 row pitch: stride 17 mod 64 -> conflict-free
#define WPAD    17      // W2 / B1 row pitch: stride 17 mod 64 -> conflict-free

// ---------------------------------------------------------------------------
// Pass 1: per (batch, chunk) partial of v[16][4] = (theta_w@x+b)·(phi_w@x+b)^T
// summed over the chunk's 256 pixels. Deterministic shuffle-tree reduction
// (no float atomics -> bitwise-reproducible across graph replays).
// ---------------------------------------------------------------------------
__global__ __launch_bounds__(256) void spd_pass1(
    const float* __restrict__ x, const float* __restrict__ theta_w,
    const float* __restrict__ theta_b, const float* __restrict__ phi_w,
    const float* __restrict__ phi_b, float* __restrict__ vpart) {
  __shared__ float tw[NNODE * CCH];
  __shared__ float pw[MIDD * CCH];
  __shared__ float wsum[8 * 64];
  const int tid = threadIdx.x;
  const int b = blockIdx.x >> 4;
  const int chunk = blockIdx.x & 15;
  const int l = chunk * CHUNKPX + tid;

  for (int i = tid; i < NNODE * CCH; i += 256) tw[i] = theta_w[i];
  for (int i = tid; i < MIDD * CCH; i += 256) pw[i] = phi_w[i];
  __syncthreads();

  float bm[NNODE];
  float xr[MIDD];
#pragma unroll
  for (int n = 0; n < NNODE; ++n) bm[n] = 0.f;
#pragma unroll
  for (int m = 0; m < MIDD; ++m) xr[m] = 0.f;

  const float* xp = x + (size_t)b * CCH * LPIX + l;
  for (int c = 0; c < CCH; ++c) {
    const float xv = xp[(size_t)c * LPIX];   // coalesced across threads
#pragma unroll
    for (int n = 0; n < NNODE; ++n) bm[n] = fmaf(tw[n * CCH + c], xv, bm[n]);
#pragma unroll
    for (int m = 0; m < MIDD; ++m) xr[m] = fmaf(pw[m * CCH + c], xv, xr[m]);
  }
#pragma unroll
  for (int n = 0; n < NNODE; ++n) bm[n] += theta_b[n];
#pragma unroll
  for (int m = 0; m < MIDD; ++m) xr[m] += phi_b[m];

  const int wave = tid >> 5, lane = tid & 31;
#pragma unroll
  for (int n = 0; n < NNODE; ++n) {
#pragma unroll
    for (int m = 0; m < MIDD; ++m) {
      float p = bm[n] * xr[m];
#pragma unroll
      for (int off = 16; off > 0; off >>= 1) p += __shfl_down(p, off, 32);
      if (lane == 0) wsum[wave * 64 + n * MIDD + m] = p;
    }
  }
  __syncthreads();
  if (tid < 64) {
    float s = 0.f;
#pragma unroll
    for (int w = 0; w < 8; ++w) s += wsum[w * 64 + tid];
    vpart[(size_t)blockIdx.x * 64 + tid] = s;
  }
}

// ---------------------------------------------------------------------------
// Pass 2: per batch: sum chunk partials -> v; hgc = g1_w@v + g1_b;
// z[n,p] = sum_m g2_w[p,m]*hgc[n,m] + g2_b[p]; W2 = phi_inv_w @ z^T  [256x16]
// ---------------------------------------------------------------------------
__global__ __launch_bounds__(256) void spd_pass2(
    const float* __restrict__ vpart, const float* __restrict__ g1_w,
    const float* __restrict__ g1_b, const float* __restrict__ g2_w,
    const float* __restrict__ g2_b, const float* __restrict__ phi_inv_w,
    float* __restrict__ W2) {
  __shared__ float vloc[64];
  __shared__ float hgc[64];
  __shared__ float zz[64];
  const int b = blockIdx.x, tid = threadIdx.x;
  if (tid < 64) {
    float s = 0.f;
    for (int ch = 0; ch < NCHUNK; ++ch)
      s += vpart[((size_t)b * NCHUNK + ch) * 64 + tid];
    vloc[tid] = s;                       // v[n][m] at n*4+m
  }
  __syncthreads();
  if (tid < 64) {
    const int o = tid >> 2, m = tid & 3;
    float s = g1_b[o];
#pragma unroll
    for (int n = 0; n < NNODE; ++n)
      s = fmaf(g1_w[o * NNODE + n], vloc[n * MIDD + m], s);
    hgc[tid] = s;
  }
  __syncthreads();
  if (tid < 64) {
    const int n = tid >> 2, p = tid & 3;
    float s = g2_b[p];
#pragma unroll
    for (int m = 0; m < MIDD; ++m)
      s = fmaf(g2_w[p * MIDD + m], hgc[n * MIDD + m], s);
    zz[tid] = s;                         // z[n][p]
  }
  __syncthreads();
  const int c = tid;                     // 0..255
#pragma unroll
  for (int k = 0; k < NNODE; ++k) {
    float s = 0.f;
#pragma unroll
    for (int m = 0; m < MIDD; ++m)
      s = fmaf(phi_inv_w[c * MIDD + m], zz[k * MIDD + m], s);
    W2[((size_t)b * CCH + c) * NNODE + k] = s;
  }
}

// ---------------------------------------------------------------------------
// Pass 3: per (batch, 64-pixel tile): two chained V_WMMA_F32_16X16X4_F32
// GEMMs.  Stage A: B1[16x16] = theta1_w[16x256] @ x_tile (K=256, 64 WMMA,
// split over TWO accumulator chains for ILP).  Stage B: x_res[256x16] =
// W2[256x16] @ B1 (K=16, 64 WMMA) + bias + residual, tanh, store.
// 4 waves/block, one 16-pixel column group per wave.
//
// f32 WMMA operand layout (ISA 7.12.2): A 16x4: lane l holds
// A[l&15][k+2*(l>>4)] in v0 and [..+1] in v1; B 4x16 mirrors with N across
// lanes; C/D: vgpr r = rows r (lanes 0-15) and r+8 (lanes 16-31).
// ---------------------------------------------------------------------------
__global__ __launch_bounds__(128) void spd_pass3(
    const float* __restrict__ x, const float* __restrict__ theta1_w,
    const float* __restrict__ theta1_b, const float* __restrict__ phi_inv_b,
    const float* __restrict__ W2, float* __restrict__ out) {
  __shared__ float xs[CCH * XPAD];          // x tile [256][64] (+pad)
  __shared__ float t1[NNODE * TPAD];        // theta1_w [16][256] (+pad)
  __shared__ float w2[CCH * WPAD];          // W2_b [256][16] (+pad)
  __shared__ float pib[CCH];                // phi_inv_b
  __shared__ float b1s[4][NNODE * WPAD];    // per-wave B1 staging [16][16](+pad)

  const int tid = threadIdx.x;
  const int b = blockIdx.y;
  const int l0 = blockIdx.x * TILEPX;

  // ---- x tile: async global->LDS B128 copies (ASYNCcnt-tracked) ----
  {
    const float* xbase = x + (size_t)b * CCH * LPIX + l0;
    for (int i = tid; i < CCH * TILEPX / 4; i += 128) {
      const int c = i >> 4;                      // channel row
      const int q = i & 15;                      // float4 within row
      const unsigned goff = (unsigned)(((size_t)c * LPIX + q * 4) * 4u);
      const unsigned lds = (unsigned)(size_t)&xs[c * XPAD + q * 4];
      asm volatile("global_load_async_to_lds_b128 %0, %1, %2"
                   :: "v"(lds), "v"(goff), "s"(xbase) : "memory");
    }
  }
  // ---- small operand tiles via regular loads ----
  for (int i = tid; i < NNODE * CCH; i += 128) {
    const int r = i >> 8, cc = i & 255;
    t1[r * TPAD + cc] = theta1_w[i];
  }
  for (int i = tid; i < CCH * NNODE; i += 128) {
    const int c = i >> 4, k = i & 15;
    w2[c * WPAD + k] = W2[((size_t)b * CCH + c) * NNODE + k];
  }
  for (int i = tid; i < CCH; i += 128) pib[i] = phi_inv_b[i];
  asm volatile("s_wait_asynccnt 0" ::: "memory");
  __syncthreads();

  const int wave = tid >> 5, lane = tid & 31;
  const int row = lane & 15, hi = lane >> 4;   // hi selects K {0,1} vs {2,3}
  const int pg = wave * 16;                    // pixel group within tile

  // -------- Stage A: B1 = theta1_w @ x_tile + theta1_b --------
  // Two independent accumulator chains (even/odd K-step) for 2x WMMA ILP.
  v8f acc0 = {}, acc1 = {};
  for (int k = 0; k < CCH; k += 8) {
    v2f a, bb;
    a.x = t1[row * TPAD + k + 2 * hi];
    a.y = t1[row * TPAD + k + 2 * hi + 1];
    bb.x = xs[(k + 2 * hi) * XPAD + pg + row];
    bb.y = xs[(k + 2 * hi + 1) * XPAD + pg + row];
    acc0 = __builtin_amdgcn_wmma_f32_16x16x4_f32(
        false, a, false, bb, (short)0, acc0, false, false);
    a.x = t1[row * TPAD + k + 4 + 2 * hi];
    a.y = t1[row * TPAD + k + 4 + 2 * hi + 1];
    bb.x = xs[(k + 4 + 2 * hi) * XPAD + pg + row];
    bb.y = xs[(k + 5 + 2 * hi) * XPAD + pg + row];
    acc1 = __builtin_amdgcn_wmma_f32_16x16x4_f32(
        false, a, false, bb, (short)0, acc1, false, false);
  }
#pragma unroll
  for (int r = 0; r < 8; ++r)
    b1s[wave][(r + 8 * hi) * WPAD + row] =
        acc0[r] + acc1[r] + theta1_b[r + 8 * hi];
  __syncthreads();

  // -------- Stage B: x_res = W2 @ B1; out = tanh(x_res + pib + x) --------
#pragma unroll 2
  for (int mt = 0; mt < CCH / NNODE; ++mt) {
    const int c0 = mt * 16;
    v8f acc2 = {};
#pragma unroll
    for (int kk = 0; kk < 4; ++kk) {
      v2f a, bb;
      a.x = w2[(c0 + row) * WPAD + 4 * kk + 2 * hi];
      a.y = w2[(c0 + row) * WPAD + 4 * kk + 2 * hi + 1];
      bb.x = b1s[wave][(4 * kk + 2 * hi) * WPAD + row];
      bb.y = b1s[wave][(4 * kk + 2 * hi + 1) * WPAD + row];
      acc2 = __builtin_amdgcn_wmma_f32_16x16x4_f32(
          false, a, false, bb, (short)0, acc2, false, false);
    }
#pragma unroll
    for (int r = 0; r < 8; ++r) {
      const int ch = c0 + r + 8 * hi;
      const int px = pg + row;
      const float val = acc2[r] + pib[ch] + xs[ch * XPAD + px];
      out[(size_t)(b * CCH + ch) * LPIX + l0 + px] = tanhf(val);
    }
  }
}

// ---------------------------------------------------------------------------
extern "C" void kernel_launch(void* const* d_in, const int* in_sizes, int n_in,
                              void* d_out, int out_size, void* d_ws,
                              size_t ws_size, hipStream_t stream) {
  (void)in_sizes; (void)n_in; (void)out_size; (void)ws_size;
  const float* x         = (const float*)d_in[0];
  const float* phi_w     = (const float*)d_in[1];
  const float* phi_b     = (const float*)d_in[2];
  const float* theta_w   = (const float*)d_in[3];
  const float* theta_b   = (const float*)d_in[4];
  const float* theta1_w  = (const float*)d_in[5];
  const float* theta1_b  = (const float*)d_in[6];
  const float* g1_w      = (const float*)d_in[7];
  const float* g1_b      = (const float*)d_in[8];
  const float* g2_w      = (const float*)d_in[9];
  const float* g2_b      = (const float*)d_in[10];
  const float* phi_inv_w = (const float*)d_in[11];
  const float* phi_inv_b = (const float*)d_in[12];
  float* out = (float*)d_out;

  // workspace: vpart [32*16][64] floats, then W2 [32][256][16] floats (~640KB)
  float* vpart = (float*)d_ws;
  float* W2 = vpart + BATCH * NCHUNK * 64;

  spd_pass1<<<dim3(BATCH * NCHUNK), dim3(256), 0, stream>>>(
      x, theta_w, theta_b, phi_w, phi_b, vpart);
  spd_pass2<<<dim3(BATCH), dim3(256), 0, stream>>>(
      vpart, g1_w, g1_b, g2_w, g2_b, phi_inv_w, W2);
  spd_pass3<<<dim3(LPIX / TILEPX, BATCH), dim3(128), 0, stream>>>(
      x, theta1_w, theta1_b, phi_inv_b, W2, out);
}